// WaveletScatter1d_8108898254980
// MI455X (gfx1250) — compile-verified
//
#include <hip/hip_runtime.h>
#include <math.h>

// ---------------- problem constants ----------------
#define T_LEN   16384
#define N1R     128             // radix: 16384 = 128 x 128
#define PK      130             // LDS plane row pitch in floats (padded)
#define PLANE   (N1R * PK)      // floats per LDS plane
#define NB      16              // batch
#define NW1     64              // first-order wavelets
#define NPAIR   224             // valid (j1,j2) pairs
#define OUT_LEN 64              // T / 2^J
#define INV_N   (1.0f/16384.0f)
#define TWO_PI  6.28318530717958647692f

typedef float v2f __attribute__((ext_vector_type(2)));
typedef float v8f __attribute__((ext_vector_type(8)));

// Explicit global (address_space(1)) accessors: guarantees global_load/store
// selection (flat_* would also tick DScnt and fight the LDS pipe).
typedef const v2f __attribute__((address_space(1)))* gv2cp;
typedef v2f __attribute__((address_space(1)))* gv2p;

static __device__ __forceinline__ v2f ldg2(const float* p) {
  return *(gv2cp)(unsigned long long)p;
}
static __device__ __forceinline__ void stg2(float* p, v2f v) {
  *(gv2p)(unsigned long long)p = v;
}

// f32 WMMA 16x16x4: A,B are 16x4 / 4x16 f32 (2 VGPRs each), C/D 16x16 f32 (8 VGPRs).
static __device__ __forceinline__ v8f wmma4(v2f a, v2f b, v8f c) {
  return __builtin_amdgcn_wmma_f32_16x16x4_f32(false, a, false, b, (short)0, c,
                                               false, false);
}

// ---------------------------------------------------------------------------
// One 128x128x128 complex GEMM step of the four-step FFT (SoA planes).
//   Logical math: out(m,n) = sum_k X[k][m] * F[k][n],  F symmetric DFT matrix.
//   Input planes hold X as plane[m*PK + k]  (A-operand: one ds_load_b64/lane).
//   F planes packed K-pair-interleaved: F[(k>>1)*256 + n*2 + (k&1)]
//     (B-operand: one coalesced global_load_b64/lane, L0-resident).
// TW    : fuse twiddle Tw[m*128+n] into the result (step A).
// MODE 0: store out(m,n) at plane[n*PK+m]   (K-major input for step C)
// MODE 1: store out(m,n) at plane[m*PK+n]   (= linear-k layout, k = m + 128n)
// MODE 2: like MODE 1 but store {|z|/N, 0}  (iFFT -> modulus chain boundary)
// 8 waves; wave w owns output column strip n in [16w,16w+16); loops 8 row tiles.
// ---------------------------------------------------------------------------
template <bool TW, int MODE>
static __device__ void cfft_pass(const float* __restrict__ ReIn,
                                 const float* __restrict__ ImIn,
                                 float* __restrict__ ReOut,
                                 float* __restrict__ ImOut,
                                 const float* __restrict__ FreP,
                                 const float* __restrict__ FimP,
                                 const float* __restrict__ Tw) {
  const int lane  = threadIdx.x & 31;
  const int wv    = threadIdx.x >> 5;
  const int lh    = lane >> 4;     // half-wave: selects K pair / M offset
  const int ll    = lane & 15;
  const int tileN = wv * 16;
  const int nB    = tileN + ll;    // B/D column owned by this lane

  for (int tm = 0; tm < 8; ++tm) {
    const int tileM = tm * 16;
    v8f aRR = {}, aII = {}, aRI = {}, aIR = {};
    const int rowA = tileM + ll;   // A-operand row (same for both lane halves)
    for (int kk = 0; kk < 128; kk += 4) {
      const int kA = kk + 2 * lh;  // lanes 0-15: K=kk,kk+1  lanes 16-31: kk+2,kk+3
      const v2f Ar = *(const v2f*)(ReIn + rowA * PK + kA);
      const v2f Ai = *(const v2f*)(ImIn + rowA * PK + kA);
      const int fo = (kA >> 1) * 256 + nB * 2;
      const v2f Br = ldg2(FreP + fo);
      const v2f Bi = ldg2(FimP + fo);
      aRR = wmma4(Ar, Br, aRR);
      aII = wmma4(Ai, Bi, aII);
      aRI = wmma4(Ar, Bi, aRI);
      aIR = wmma4(Ai, Br, aIR);
    }
#pragma unroll
    for (int r = 0; r < 8; ++r) {
      float re = aRR[r] - aII[r];
      float im = aRI[r] + aIR[r];
      const int m = tileM + r + 8 * lh;  // D layout: VGPR r, lanes16-31 -> M+8
      const int n = nB;
      if (TW) {
        const v2f w = ldg2(Tw + 2 * ((m << 7) + n));
        const float tr = re * w.x - im * w.y;
        im = re * w.y + im * w.x;
        re = tr;
      }
      if (MODE == 0) {
        ReOut[n * PK + m] = re;
        ImOut[n * PK + m] = im;
      } else if (MODE == 1) {
        ReOut[m * PK + n] = re;
        ImOut[m * PK + n] = im;
      } else {
        ReOut[m * PK + n] = sqrtf(re * re + im * im) * INV_N;
        ImOut[m * PK + n] = 0.0f;
      }
    }
  }
}

// Full 16384-pt transform. Input/result in planes A with linear-k layout:
// element k at plane[(k&127)*PK + (k>>7)].
// FINAL=1: result spectrum, linear layout.  FINAL=0: chain boundary ->
// planes A hold {|ifft|/N, 0} laid out as the next forward-FFT input.
template <int FINAL>
static __device__ void fft16384(float* ReA, float* ImA, float* ReB, float* ImB,
                                const float* __restrict__ FreP,
                                const float* __restrict__ FimP,
                                const float* __restrict__ Tw) {
  cfft_pass<true, 0>(ReA, ImA, ReB, ImB, FreP, FimP, Tw);
  __syncthreads();
  if (FINAL)
    cfft_pass<false, 1>(ReB, ImB, ReA, ImA, FreP, FimP, nullptr);
  else
    cfft_pass<false, 2>(ReB, ImB, ReA, ImA, FreP, FimP, nullptr);
  __syncthreads();
}

// Lowpass (phi) + fold to 64 bins + 64-pt inverse DFT (real part), scaled 1/T.
// Planes hold spectrum in linear layout; scratch = B planes (>=256 float2).
static __device__ void fold_output(const float* __restrict__ Re,
                                   const float* __restrict__ Im,
                                   const float* __restrict__ phi,
                                   float* __restrict__ outp,
                                   float2* __restrict__ scratch) {
  __syncthreads();
  const int t = threadIdx.x;
  const int m = t & 63;
  const int q = t >> 6;
  float sr = 0.0f, si = 0.0f;
  for (int rr = 0; rr < 64; ++rr) {
    const int k = m + 64 * (q + 4 * rr);
    const int a = (k & 127) * PK + (k >> 7);
    const float f = phi[k];
    sr += Re[a] * f;
    si += Im[a] * f;
  }
  scratch[t] = make_float2(sr, si);
  __syncthreads();
  if (t < 64) {
    float acc = 0.0f;
    for (int mm = 0; mm < 64; ++mm) {
      const float2 g0 = scratch[mm];
      const float2 g1 = scratch[mm + 64];
      const float2 g2 = scratch[mm + 128];
      const float2 g3 = scratch[mm + 192];
      const float gr = g0.x + g1.x + g2.x + g3.x;
      const float gi = g0.y + g1.y + g2.y + g3.y;
      const int pp = (mm * t) & 63;
      float s, c;
      __sincosf((float)pp * (TWO_PI / 64.0f), &s, &c);
      acc += gr * c - gi * s;  // Re{ G * e^{+i theta} }
    }
    outp[t] = acc * INV_N;
  }
  __syncthreads();
}

// Load spectrum * real filter into planes (linear-k layout).
static __device__ void load_filtered(const float* __restrict__ src /*float2*/,
                                     const float* __restrict__ filt,
                                     float* __restrict__ Re,
                                     float* __restrict__ Im) {
  for (int k = threadIdx.x; k < T_LEN; k += 256) {
    const v2f v = ldg2(src + 2 * k);
    const float f = filt[k];
    const int a = (k & 127) * PK + (k >> 7);
    Re[a] = v.x * f;
    Im[a] = v.y * f;
  }
}

// ---------------- kernels ----------------

// Build K-pair-packed 128-pt DFT planes (fwd/bwd) and N=16384 twiddles (fwd/bwd).
__global__ __launch_bounds__(256) void init_tables(
    float* __restrict__ Ff_re, float* __restrict__ Ff_im,
    float* __restrict__ Fb_re, float* __restrict__ Fb_im,
    float* __restrict__ Twf, float* __restrict__ Twb) {
  const int i = blockIdx.x * blockDim.x + threadIdx.x;
  if (i >= 128 * 128) return;
  const int k = i >> 7, n = i & 127;
  {
    const int p = (k * n) & 127;
    float s, c;
    __sincosf((float)p * (TWO_PI / 128.0f), &s, &c);
    const int pos = (k >> 1) * 256 + n * 2 + (k & 1);
    Ff_re[pos] = c;  Ff_im[pos] = -s;   // e^{-2pi i kn/128}
    Fb_re[pos] = c;  Fb_im[pos] = s;    // e^{+2pi i kn/128}
  }
  {
    const int p = (k * n) & 16383;
    float s, c;
    __sincosf((float)p * (TWO_PI / 16384.0f), &s, &c);
    stg2(Twf + 2 * i, (v2f){c, -s});    // e^{-2pi i kn/N}
    stg2(Twb + 2 * i, (v2f){c, s});     // e^{+2pi i kn/N}
  }
}

// xh = FFT(x) per batch row; also S0 = fold(xh * phi).
__global__ __launch_bounds__(256) void k_fft_x(
    const float* __restrict__ x, const float* __restrict__ phi,
    const float* __restrict__ Ff_re, const float* __restrict__ Ff_im,
    const float* __restrict__ Twf, float* __restrict__ xh /*float2*/,
    float* __restrict__ out /* S0 base */) {
  extern __shared__ float smem[];
  float* ReA = smem;
  float* ImA = smem + PLANE;
  float* ReB = smem + 2 * PLANE;
  float* ImB = smem + 3 * PLANE;
  const int b = blockIdx.x;
  const float* xr = x + (size_t)b * T_LEN;
  for (int k = threadIdx.x * 64; k < T_LEN; k += 256 * 64)
    __builtin_prefetch(phi + k, 0, 1);
  for (int k = threadIdx.x; k < T_LEN; k += 256) {
    const int a = (k & 127) * PK + (k >> 7);
    ReA[a] = xr[k];
    ImA[a] = 0.0f;
  }
  __syncthreads();
  fft16384<1>(ReA, ImA, ReB, ImB, Ff_re, Ff_im, Twf);
  float* xo = xh + (size_t)b * T_LEN * 2;
  for (int k = threadIdx.x; k < T_LEN; k += 256) {
    const int a = (k & 127) * PK + (k >> 7);
    stg2(xo + 2 * k, (v2f){ReA[a], ImA[a]});
  }
  fold_output(ReA, ImA, phi, out + (size_t)b * OUT_LEN, (float2*)ReB);
}

// Order 1: per (b,j): Y = xh*psi1_j -> iFFT -> |.| -> FFT -> fold(phi) -> S1.
__global__ __launch_bounds__(256) void k_order1(
    const float* __restrict__ xh /*float2*/, const float* __restrict__ psi1,
    const float* __restrict__ phi,
    const float* __restrict__ Ff_re, const float* __restrict__ Ff_im,
    const float* __restrict__ Twf,
    const float* __restrict__ Fb_re, const float* __restrict__ Fb_im,
    const float* __restrict__ Twb, float* __restrict__ out /* S1 base */) {
  extern __shared__ float smem[];
  float* ReA = smem;
  float* ImA = smem + PLANE;
  float* ReB = smem + 2 * PLANE;
  float* ImB = smem + 3 * PLANE;
  const int b = blockIdx.x >> 6;
  const int j = blockIdx.x & 63;
  for (int k = threadIdx.x * 64; k < T_LEN; k += 256 * 64)
    __builtin_prefetch(phi + k, 0, 1);
  load_filtered(xh + (size_t)b * T_LEN * 2, psi1 + (size_t)j * T_LEN, ReA, ImA);
  __syncthreads();
  fft16384<0>(ReA, ImA, ReB, ImB, Fb_re, Fb_im, Twb);  // U1 = |ifft|
  fft16384<1>(ReA, ImA, ReB, ImB, Ff_re, Ff_im, Twf);  // U1h linear
  fold_output(ReA, ImA, phi, out + ((size_t)(b * NW1 + j)) * OUT_LEN,
              (float2*)ReB);
}

// Order 2: per (b,p): recompute U1h for j1, filter psi2_j2, iFFT, |.|, FFT, fold.
__global__ __launch_bounds__(256) void k_order2(
    const float* __restrict__ xh /*float2*/, const float* __restrict__ psi1,
    const float* __restrict__ psi2, const float* __restrict__ phi,
    const int* __restrict__ pj1, const int* __restrict__ pj2,
    const float* __restrict__ Ff_re, const float* __restrict__ Ff_im,
    const float* __restrict__ Twf,
    const float* __restrict__ Fb_re, const float* __restrict__ Fb_im,
    const float* __restrict__ Twb, float* __restrict__ out /* S2 base */) {
  extern __shared__ float smem[];
  float* ReA = smem;
  float* ImA = smem + PLANE;
  float* ReB = smem + 2 * PLANE;
  float* ImB = smem + 3 * PLANE;
  const int blk = blockIdx.x;
  const int b = blk / NPAIR;
  const int p = blk - b * NPAIR;
  const int j1 = pj1[p];
  const int j2 = pj2[p];
  const float* f2 = psi2 + (size_t)j2 * T_LEN;
  for (int k = threadIdx.x * 64; k < T_LEN; k += 256 * 64) {
    __builtin_prefetch(phi + k, 0, 1);
    __builtin_prefetch(f2 + k, 0, 1);
  }
  load_filtered(xh + (size_t)b * T_LEN * 2, psi1 + (size_t)j1 * T_LEN, ReA,
                ImA);
  __syncthreads();
  fft16384<0>(ReA, ImA, ReB, ImB, Fb_re, Fb_im, Twb);  // U1 = |ifft|
  fft16384<1>(ReA, ImA, ReB, ImB, Ff_re, Ff_im, Twf);  // U1h linear
  for (int k = threadIdx.x; k < T_LEN; k += 256) {
    const int a = (k & 127) * PK + (k >> 7);
    const float f = f2[k];
    ReA[a] *= f;
    ImA[a] *= f;
  }
  __syncthreads();
  fft16384<0>(ReA, ImA, ReB, ImB, Fb_re, Fb_im, Twb);  // U2 = |ifft|
  fft16384<1>(ReA, ImA, ReB, ImB, Ff_re, Ff_im, Twf);  // U2h linear
  fold_output(ReA, ImA, phi, out + ((size_t)(b * NPAIR + p)) * OUT_LEN,
              (float2*)ReB);
}

// ---------------- launch ----------------
extern "C" void kernel_launch(void* const* d_in, const int* in_sizes, int n_in,
                              void* d_out, int out_size, void* d_ws,
                              size_t ws_size, hipStream_t stream) {
  const float* x    = (const float*)d_in[0];
  const float* psi1 = (const float*)d_in[1];
  const float* psi2 = (const float*)d_in[2];
  const float* phi  = (const float*)d_in[3];
  const int*   pj1  = (const int*)d_in[4];
  const int*   pj2  = (const int*)d_in[5];
  float* out = (float*)d_out;

  char* ws = (char*)d_ws;
  float* Ff_re = (float*)(ws + 0);         // 64KB each
  float* Ff_im = (float*)(ws + 65536);
  float* Fb_re = (float*)(ws + 131072);
  float* Fb_im = (float*)(ws + 196608);
  float* Twf   = (float*)(ws + 262144);    // 128KB each (float2 pairs)
  float* Twb   = (float*)(ws + 393216);
  float* xh    = (float*)(ws + 524288);    // B*T*8 = 2MB (float2 pairs)

  const size_t smem = (size_t)4 * PLANE * sizeof(float);  // 266240 B

  init_tables<<<64, 256, 0, stream>>>(Ff_re, Ff_im, Fb_re, Fb_im, Twf, Twb);
  k_fft_x<<<NB, 256, smem, stream>>>(x, phi, Ff_re, Ff_im, Twf, xh, out);
  k_order1<<<NB * NW1, 256, smem, stream>>>(xh, psi1, phi, Ff_re, Ff_im, Twf,
                                            Fb_re, Fb_im, Twb,
                                            out + NB * OUT_LEN);
  k_order2<<<NB * NPAIR, 256, smem, stream>>>(
      xh, psi1, psi2, phi, pj1, pj2, Ff_re, Ff_im, Twf, Fb_re, Fb_im, Twb,
      out + NB * OUT_LEN + NB * NW1 * OUT_LEN);
}